// AxialAttention_74062416052441
// MI455X (gfx1250) — compile-verified
//
#include <hip/hip_runtime.h>
#include <hip/hip_bf16.h>

typedef __attribute__((ext_vector_type(2))) float v2f;
typedef __attribute__((ext_vector_type(8))) float v8f;
typedef int gv4i __attribute__((vector_size(16)));   // matches builtin pointee

#define NB   8
#define CIN  128
#define HH   56
#define WW   56
#define SS   (HH*WW)          // 3136
#define COQK 64               // q,k channels each
#define COV  128
#define CALL 256              // 64 q + 64 k + 128 v
#define GG   8
#define GPH  8                // gp/2 (q,k planes per group)
#define GPV  16               // v planes per group
#define EPSB 1e-5f

// CDNA5 async global->LDS DMA (ASYNCcnt) when the toolchain declares it.
#define AS1 __attribute__((address_space(1)))
#define AS3 __attribute__((address_space(3)))
#if defined(__AMDGCN__) && defined(__has_builtin)
#  if __has_builtin(__builtin_amdgcn_global_load_async_to_lds_b128) && \
      __has_builtin(__builtin_amdgcn_global_load_async_to_lds_b32)
#    define HAVE_ASYNC_LDS 1
#  endif
#endif
#ifndef HAVE_ASYNC_LDS
#  define HAVE_ASYNC_LDS 0
#endif

// ---------------- workspace layout (floats) ----------------
static constexpr size_t OFF_QKV = 0;
static constexpr size_t OFF_QK  = OFF_QKV + (size_t)NB*CALL*SS;
static constexpr size_t OFF_QR  = OFF_QK  + (size_t)NB*GG*WW*HH*HH;
static constexpr size_t OFF_KR  = OFF_QR  + (size_t)NB*GG*HH*WW;
static constexpr size_t OFF_SV  = OFF_KR  + (size_t)NB*GG*HH*WW;
static constexpr size_t OFF_SVE = OFF_SV  + (size_t)NB*COV*SS;
static constexpr size_t OFF_STQ = OFF_SVE + (size_t)NB*COV*SS;
static constexpr size_t OFF_STG = OFF_STQ + 512;
static constexpr size_t OFF_SSV = OFF_STG + 48;
static constexpr size_t OFF_SSE = OFF_SSV + 256;

__device__ __forceinline__ v8f wmma_f32(v2f a, v2f b, v8f c) {
  return __builtin_amdgcn_wmma_f32_16x16x4_f32(false, a, false, b, (short)0, c,
                                               false, false);
}

// ============ 1) fused projection GEMM: qkv_raw = [wq;wk;wv] @ x ============
// One block = (b, 64-wide spatial slice). X tile staged once in LDS (async DMA
// when available); 8 waves x 2 o-tiles x 4 N-subtiles, 256 WMMA per wave.
__global__ __launch_bounds__(256) void proj_gemm(
    const float* __restrict__ x, const float* __restrict__ wq,
    const float* __restrict__ wk, const float* __restrict__ wv,
    float* __restrict__ qkv) {
  __shared__ __align__(16) float Xs[CIN * 64];    // [k][s], 32 KB
  const int t  = threadIdx.x;
  const int b  = blockIdx.x / 49;
  const int s0 = (blockIdx.x % 49) * 64;
  const float* Xb = x + (size_t)b * CIN * SS;

#if HAVE_ASYNC_LDS
  for (int chunk = t; chunk < CIN * 16; chunk += 256) {   // 16B chunks
    int k = chunk >> 4, c4 = (chunk & 15) << 2;
    __builtin_amdgcn_global_load_async_to_lds_b128(
        (AS1 gv4i*)(Xb + (size_t)k * SS + s0 + c4),
        (AS3 gv4i*)(Xs + k * 64 + c4), 0, 0);
  }
  asm volatile("s_wait_asynccnt 0x0" ::: "memory");
#else
  for (int chunk = t; chunk < CIN * 16; chunk += 256) {
    int k = chunk >> 4, c4 = (chunk & 15) << 2;
    *(float4*)(Xs + k * 64 + c4) =
        *(const float4*)(Xb + (size_t)k * SS + s0 + c4);
  }
#endif
  __syncthreads();

  const int lane = t & 31, wid = t >> 5, half = lane >> 4, l = lane & 15;
  for (int task = wid; task < 16; task += 8) {    // 16 o-tiles of 16 rows
    const int o0 = task * 16;
    const int orow = o0 + l;
    const float* Wrow;
    if (orow < COQK)          Wrow = wq + (size_t)orow * CIN;
    else if (orow < 2*COQK)   Wrow = wk + (size_t)(orow - COQK) * CIN;
    else                      Wrow = wv + (size_t)(orow - 2*COQK) * CIN;

    v8f a0 = {}, a1 = {}, a2 = {}, a3 = {};
#pragma unroll 4
    for (int k = 0; k < CIN; k += 4) {
      v2f a = *(const v2f*)(Wrow + k + 2 * half); // A: lanes0-15 K{0,1}, 16-31 K{2,3}
      const float* B0 = Xs + (k + 2 * half) * 64 + l;
      const float* B1 = Xs + (k + 2 * half + 1) * 64 + l;
      v2f b0 = {B0[0],  B1[0]};
      v2f b1 = {B0[16], B1[16]};
      v2f b2 = {B0[32], B1[32]};
      v2f b3 = {B0[48], B1[48]};
      a0 = wmma_f32(a, b0, a0);
      a1 = wmma_f32(a, b1, a1);
      a2 = wmma_f32(a, b2, a2);
      a3 = wmma_f32(a, b3, a3);
    }
    float* dst = qkv + (size_t)b * CALL * SS + s0 + l;
#pragma unroll
    for (int r = 0; r < 8; ++r) {
      int o = o0 + r + 8 * half;                  // D: M = r + 8*(lane/16)
      float* dr = dst + (size_t)o * SS;
      dr[0] = a0[r]; dr[16] = a1[r]; dr[32] = a2[r]; dr[48] = a3[r];
    }
  }
}

// ============ 2) BN stats for the 256 projection channels ============
__global__ __launch_bounds__(256) void stats_qkv_kernel(
    const float* __restrict__ qkv, const float* g_q, const float* b_q,
    const float* g_k, const float* b_k, const float* g_v, const float* b_v,
    float* __restrict__ st) {
  __shared__ float s1[256], s2[256];
  const int ch = blockIdx.x, t = threadIdx.x;
  float a = 0.f, q = 0.f;
  for (int e = t; e < NB * SS; e += 256) {
    int bb = e / SS, s = e % SS;
    float v = qkv[((size_t)bb * CALL + ch) * SS + s];
    a += v; q += v * v;
  }
  s1[t] = a; s2[t] = q; __syncthreads();
  for (int off = 128; off > 0; off >>= 1) {
    if (t < off) { s1[t] += s1[t + off]; s2[t] += s2[t + off]; }
    __syncthreads();
  }
  if (t == 0) {
    const float cnt = (float)(NB * SS);
    float m = s1[0] / cnt;
    float var = s2[0] / cnt - m * m;
    float rstd = rsqrtf(var + EPSB);
    float gam, bet;
    if (ch < COQK)           { gam = g_q[ch];            bet = b_q[ch]; }
    else if (ch < 2 * COQK)  { gam = g_k[ch - COQK];     bet = b_k[ch - COQK]; }
    else                     { gam = g_v[ch - 2*COQK];   bet = b_v[ch - 2*COQK]; }
    float sc = gam * rstd;
    st[ch] = sc; st[256 + ch] = bet - m * sc;
  }
}

// ============ 3) qk = qhat^T khat per (b,g,w) + qr/kr bases ============
__global__ __launch_bounds__(256) void qk_kernel(
    const float* __restrict__ qkv, const float* __restrict__ st,
    const float* __restrict__ q_rel, const float* __restrict__ k_rel,
    float* __restrict__ qkraw, float* __restrict__ qrb, float* __restrict__ krb) {
  __shared__ float qh[GPH * 64];   // [c][i], i padded to 64
  __shared__ float kh[GPH * 64];
  const int t  = threadIdx.x;
  const int bg = blockIdx.x / WW, w = blockIdx.x % WW;
  const int b  = bg >> 3, g = bg & 7;

  for (int e = t; e < GPH * 64; e += 256) {
    int c = e >> 6, i = e & 63;
    float vq = 0.f, vk = 0.f;
    if (i < HH) {
      int oq = g * GPH + c, ok = COQK + g * GPH + c;
      vq = st[oq] * qkv[((size_t)b * CALL + oq) * SS + i * WW + w] + st[256 + oq];
      vk = st[ok] * qkv[((size_t)b * CALL + ok) * SS + i * WW + w] + st[256 + ok];
    }
    qh[e] = vq; kh[e] = vk;
  }
  __syncthreads();

  if (t < HH) {                                   // relative position bases
    float aq = 0.f, ak = 0.f;
#pragma unroll
    for (int c = 0; c < GPH; ++c) {
      aq += qh[c * 64 + t] * q_rel[c * HH + t];
      ak += kh[c * 64 + t] * k_rel[c * HH + t];
    }
    qrb[((size_t)bg * HH + t) * WW + w] = aq;
    krb[((size_t)bg * HH + t) * WW + w] = ak;
  }

  const int lane = t & 31, wid = t >> 5, half = lane >> 4, l = lane & 15;
  float* base = qkraw + (size_t)blockIdx.x * (HH * HH);
  for (int task = wid; task < 16; task += 8) {    // 4x4 tiles of 16x16
    int i0 = (task >> 2) * 16, j0 = (task & 3) * 16;
    v8f acc = {};
#pragma unroll
    for (int ks = 0; ks < GPH; ks += 4) {
      v2f a, bb;
      a.x  = qh[(ks + 2 * half) * 64 + i0 + l];
      a.y  = qh[(ks + 2 * half + 1) * 64 + i0 + l];
      bb.x = kh[(ks + 2 * half) * 64 + j0 + l];
      bb.y = kh[(ks + 2 * half + 1) * 64 + j0 + l];
      acc = wmma_f32(a, bb, acc);
    }
#pragma unroll
    for (int r = 0; r < 8; ++r) {
      int i = i0 + r + 8 * half, j = j0 + l;
      if (i < HH && j < HH) base[i * HH + j] = acc[r];
    }
  }
}

// ============ 4) per-group BN stats for qk / qr / kr ============
__global__ __launch_bounds__(256) void stats_g_kernel(
    const float* __restrict__ qkraw, const float* __restrict__ qrb,
    const float* __restrict__ krb, const float* g_qk, const float* b_qk,
    const float* g_qr, const float* b_qr, const float* g_kr, const float* b_kr,
    float* __restrict__ stg) {
  __shared__ float s1[256], s2[256];
  const int tensor = blockIdx.x >> 3, g = blockIdx.x & 7, t = threadIdx.x;
  float a = 0.f, q = 0.f;
  if (tensor == 0) {
    for (int b = 0; b < NB; ++b) {
      const float* base = qkraw + (size_t)((b * GG + g) * WW) * (HH * HH);
      for (int e = t; e < WW * HH * HH; e += 256) { float v = base[e]; a += v; q += v * v; }
    }
  } else {
    const float* src = (tensor == 1) ? qrb : krb;
    for (int b = 0; b < NB; ++b) {
      const float* base = src + (size_t)(b * GG + g) * (HH * WW);
      for (int e = t; e < HH * WW; e += 256) { float v = base[e]; a += v; q += v * v; }
    }
  }
  s1[t] = a; s2[t] = q; __syncthreads();
  for (int off = 128; off > 0; off >>= 1) {
    if (t < off) { s1[t] += s1[t + off]; s2[t] += s2[t + off]; }
    __syncthreads();
  }
  if (t == 0) {
    // broadcast along j does not change mean/var -> base stats are exact
    const float cnt = (tensor == 0) ? (float)(NB * WW * HH * HH) : (float)(NB * HH * WW);
    float m = s1[0] / cnt;
    float var = s2[0] / cnt - m * m;
    float rstd = rsqrtf(var + EPSB);
    float gam = (tensor == 0) ? g_qk[g] : (tensor == 1) ? g_qr[g] : g_kr[g];
    float bet = (tensor == 0) ? b_qk[g] : (tensor == 1) ? b_qr[g] : b_kr[g];
    float sc = gam * rstd;
    stg[tensor * 16 + g] = sc;
    stg[tensor * 16 + 8 + g] = bet - m * sc;
  }
}

// ============ 5) softmax + sv + sve per (b,g,w) ============
__global__ __launch_bounds__(256) void attn_kernel(
    const float* __restrict__ qkv, const float* __restrict__ st,
    const float* __restrict__ stg, const float* __restrict__ qkraw,
    const float* __restrict__ qrb, const float* __restrict__ krb,
    const float* __restrict__ v_rel, float* __restrict__ svraw,
    float* __restrict__ sveraw) {
  __shared__ float sim[64 * 64];     // [i][j] padded
  __shared__ float bT[64 * GPV];     // [j][c] = vhat[c][j]
  __shared__ float vrT[64 * GPV];    // [j][c] = v_rel[c][j]
  __shared__ float rowb[64], colb[64];
  const int t  = threadIdx.x;
  const int bg = blockIdx.x / WW, w = blockIdx.x % WW;
  const int b  = bg >> 3, g = bg & 7;

  const float sqk = stg[g],      hqk = stg[8 + g];
  const float sqr = stg[16 + g], hqr = stg[24 + g];
  const float skr = stg[32 + g], hkr = stg[40 + g];

  if (t < 64) {
    float rv = 0.f, cv = 0.f;
    if (t < HH) {
      size_t idx = ((size_t)bg * HH + t) * WW + w;
      rv = sqr * qrb[idx] + hqr;                  // qr: bias per row i
      cv = skr * krb[idx] + hkr;                  // kr (transposed): bias per col j
    }
    rowb[t] = rv; colb[t] = cv;
  }
  __syncthreads();

  const float* qks = qkraw + (size_t)blockIdx.x * (HH * HH);
  for (int e = t; e < 64 * 64; e += 256) {
    int i = e >> 6, j = e & 63;
    float val = -3.0e38f;
    if (i < HH && j < HH) val = sqk * qks[i * HH + j] + hqk + rowb[i] + colb[j];
    sim[e] = val;
  }
  __syncthreads();

  if (t < 64) {                                   // softmax over j per row i
    float* row = sim + t * 64;
    if (t < HH) {
      float m = -3.0e38f;
      for (int j = 0; j < HH; ++j) m = fmaxf(m, row[j]);
      float s = 0.f;
      for (int j = 0; j < HH; ++j) { float e2 = __expf(row[j] - m); row[j] = e2; s += e2; }
      float inv = 1.f / s;
      for (int j = 0; j < HH; ++j) row[j] *= inv;
      for (int j = HH; j < 64; ++j) row[j] = 0.f; // K padding contributes 0
    } else {
      for (int j = 0; j < 64; ++j) row[j] = 0.f;  // padded rows: keep finite
    }
  }

  for (int e = t; e < 64 * GPV; e += 256) {       // stage B matrices [j][c]
    int j = e >> 4, c = e & 15;
    float vv = 0.f;
    if (j < HH) {
      int ov = 2 * COQK + g * GPV + c;
      vv = st[ov] * qkv[((size_t)b * CALL + ov) * SS + j * WW + w] + st[256 + ov];
    }
    bT[e] = vv;
#if HAVE_ASYNC_LDS
    if (j < HH) {
      __builtin_amdgcn_global_load_async_to_lds_b32(
          (AS1 int*)(v_rel + c * HH + j), (AS3 int*)(vrT + e), 0, 0);
    } else {
      vrT[e] = 0.f;
    }
#else
    vrT[e] = (j < HH) ? v_rel[c * HH + j] : 0.f;
#endif
  }
#if HAVE_ASYNC_LDS
  asm volatile("s_wait_asynccnt 0x0" ::: "memory");
#endif
  __syncthreads();

  const int lane = t & 31, wid = t >> 5, half = lane >> 4, l = lane & 15;
  const int it = wid & 3, which = wid >> 2;       // 4 i-tiles x {sv, sve}
  const int i0 = it * 16;
  const float* Bm = which ? vrT : bT;
  v8f acc = {};
#pragma unroll
  for (int ks = 0; ks < 64; ks += 4) {
    v2f a = *(const v2f*)(sim + (i0 + l) * 64 + ks + 2 * half);
    v2f bb;
    bb.x = Bm[(ks + 2 * half) * GPV + l];
    bb.y = Bm[(ks + 2 * half + 1) * GPV + l];
    acc = wmma_f32(a, bb, acc);
  }
  float* dst = which ? sveraw : svraw;
#pragma unroll
  for (int r = 0; r < 8; ++r) {
    int i = i0 + r + 8 * half;                    // l < 16 == GPV channels
    if (i < HH)
      dst[((size_t)b * COV + g * GPV + l) * SS + i * WW + w] = acc[r];
  }
}

// ============ 6) BN stats for sv / sve (128 channels each) ============
__global__ __launch_bounds__(256) void stats_out_kernel(
    const float* __restrict__ svraw, const float* __restrict__ sveraw,
    const float* g_sv, const float* b_sv, const float* g_sve,
    const float* b_sve, float* __restrict__ ssv, float* __restrict__ sse) {
  __shared__ float s1[256], s2[256];
  const int which = blockIdx.x >> 7, ch = blockIdx.x & 127, t = threadIdx.x;
  const float* src = which ? sveraw : svraw;
  float a = 0.f, q = 0.f;
  for (int e = t; e < NB * SS; e += 256) {
    int bb = e / SS, s = e % SS;
    float v = src[((size_t)bb * COV + ch) * SS + s];
    a += v; q += v * v;
  }
  s1[t] = a; s2[t] = q; __syncthreads();
  for (int off = 128; off > 0; off >>= 1) {
    if (t < off) { s1[t] += s1[t + off]; s2[t] += s2[t + off]; }
    __syncthreads();
  }
  if (t == 0) {
    const float cnt = (float)(NB * SS);
    float m = s1[0] / cnt;
    float var = s2[0] / cnt - m * m;
    float rstd = rsqrtf(var + EPSB);
    float gam = which ? g_sve[ch] : g_sv[ch];
    float bet = which ? b_sve[ch] : b_sv[ch];
    float sc = gam * rstd;
    float* stp = which ? sse : ssv;
    stp[ch] = sc; stp[128 + ch] = bet - m * sc;
  }
}

// ============ 7) out = bn(sv) + bn(sve) ============
__global__ __launch_bounds__(256) void final_kernel(
    const float* __restrict__ svraw, const float* __restrict__ sveraw,
    const float* __restrict__ ssv, const float* __restrict__ sse,
    float* __restrict__ out) {
  size_t idx = (size_t)blockIdx.x * 256 + threadIdx.x;
  if (idx >= (size_t)NB * COV * SS) return;
  int ch = (int)((idx / SS) & 127);
  out[idx] = ssv[ch] * svraw[idx] + ssv[128 + ch] +
             sse[ch] * sveraw[idx] + sse[128 + ch];
}

extern "C" void kernel_launch(void* const* d_in, const int* in_sizes, int n_in,
                              void* d_out, int out_size, void* d_ws,
                              size_t ws_size, hipStream_t stream) {
  const float* x     = (const float*)d_in[0];
  const float* wq    = (const float*)d_in[1];
  const float* wk    = (const float*)d_in[2];
  const float* wv    = (const float*)d_in[3];
  const float* q_rel = (const float*)d_in[4];
  const float* k_rel = (const float*)d_in[5];
  const float* v_rel = (const float*)d_in[6];
  const float* g_q  = (const float*)d_in[7],  *b_q  = (const float*)d_in[8];
  const float* g_k  = (const float*)d_in[9],  *b_k  = (const float*)d_in[10];
  const float* g_v  = (const float*)d_in[11], *b_v  = (const float*)d_in[12];
  const float* g_qr = (const float*)d_in[13], *b_qr = (const float*)d_in[14];
  const float* g_kr = (const float*)d_in[15], *b_kr = (const float*)d_in[16];
  const float* g_qk = (const float*)d_in[17], *b_qk = (const float*)d_in[18];
  const float* g_sv = (const float*)d_in[19], *b_sv = (const float*)d_in[20];
  const float* g_sve= (const float*)d_in[21], *b_sve= (const float*)d_in[22];
  float* out = (float*)d_out;

  float* ws   = (float*)d_ws;
  float* qkv  = ws + OFF_QKV;
  float* qk   = ws + OFF_QK;
  float* qrb  = ws + OFF_QR;
  float* krb  = ws + OFF_KR;
  float* sv   = ws + OFF_SV;
  float* sve  = ws + OFF_SVE;
  float* stq  = ws + OFF_STQ;
  float* stg  = ws + OFF_STG;
  float* ssv  = ws + OFF_SSV;
  float* sse  = ws + OFF_SSE;

  // 1) projections: 392 blocks, X tile staged in LDS, 256 WMMA per wave
  proj_gemm<<<NB * 49, 256, 0, stream>>>(x, wq, wk, wv, qkv);
  // 2) q/k/v BN stats
  stats_qkv_kernel<<<256, 256, 0, stream>>>(qkv, g_q, b_q, g_k, b_k, g_v, b_v, stq);
  // 3) qk scores + qr/kr bases, one block per (b,g,w)
  qk_kernel<<<NB * GG * WW, 256, 0, stream>>>(qkv, stq, q_rel, k_rel, qk, qrb, krb);
  // 4) per-group stats for qk/qr/kr
  stats_g_kernel<<<24, 256, 0, stream>>>(qk, qrb, krb, g_qk, b_qk, g_qr, b_qr,
                                         g_kr, b_kr, stg);
  // 5) softmax + sv + sve
  attn_kernel<<<NB * GG * WW, 256, 0, stream>>>(qkv, stq, stg, qk, qrb, krb,
                                                v_rel, sv, sve);
  // 6) sv/sve BN stats
  stats_out_kernel<<<256, 256, 0, stream>>>(sv, sve, g_sv, b_sv, g_sve, b_sve,
                                            ssv, sse);
  // 7) final fused BN add
  const size_t total = (size_t)NB * COV * SS;
  final_kernel<<<(unsigned)((total + 255) / 256), 256, 0, stream>>>(sv, sve, ssv,
                                                                    sse, out);
}